// CRF_88476326297868
// MI455X (gfx1250) — compile-verified
//
#include <hip/hip_runtime.h>
#include <stdint.h>

#define LBL   64
#define NEGV  -10000.0f
#define CHUNK 64            // timesteps staged per async chunk (16 KB)
#define KBLK  512           // timesteps per backtrack block (32 KB LDS)

__device__ __forceinline__ int imin_(int a, int b) { return a < b ? a : b; }

// Packed backpointer layout: u32 word q = t/4 for tag n lives at u32 index
// q*64 + n  (byte address q*256 + n*4 + (t&3)).  For a block starting at
// tlo (multiple of 4), the block's region is contiguous at byte tlo*64.
__device__ __forceinline__ int bpLdsIdx(int i, int tag) {
    return ((i >> 2) << 8) + (tag << 2) + (i & 3);
}

// ---------------------------------------------------------------------------
// Kernel 1: sequential Viterbi forward pass. 1 workgroup, 64 threads (2 waves).
// Thread n owns next-tag n (transition row in VGPRs). v[] double-buffered in
// LDS. Features streamed HBM->LDS with CDNA5 async-to-LDS loads, 2 chunks
// ahead, gated by s_wait_asynccnt. Backpointers packed 4 steps per u32.
// ---------------------------------------------------------------------------
__global__ __launch_bounds__(64) void crf_forward(
    const float* __restrict__ feats, const float* __restrict__ trans,
    unsigned* __restrict__ bp32, float* __restrict__ outScore,
    int* __restrict__ bestOut, int T)
{
    __shared__ __align__(16) float vbuf[2][LBL];
    __shared__ __align__(16) float featbuf[2][CHUNK * LBL];
    __shared__ float term[LBL];

    const int n = threadIdx.x;   // next-tag index

    // transitions[n][0..63] -> registers
    float tr[LBL];
#pragma unroll
    for (int i = 0; i < 16; i++) {
        float4 q = ((const float4*)(trans + n * LBL))[i];
        tr[4*i+0] = q.x; tr[4*i+1] = q.y; tr[4*i+2] = q.z; tr[4*i+3] = q.w;
    }

    vbuf[0][n] = (n == 62) ? 0.0f : NEGV;   // START = 62

    const bool     w0     = (n < 32);
    const unsigned lane   = (unsigned)(n & 31);
    const uint64_t gbase  = (uint64_t)(uintptr_t)feats;
    const uint64_t fbytes = (uint64_t)T * (LBL * 4);
    const int      NC     = (T + CHUNK - 1) / CHUNK;

    // Issue one 16 KB chunk as 32 async b128 transfers (wave0 only).
    // Addresses are clamped into the valid feats range so every chunk issues
    // exactly 32 ops -> ASYNCcnt bookkeeping stays uniform.
    auto issue = [&](int c, int buf) {
        uint64_t cb    = (uint64_t)c * (uint64_t)(CHUNK * LBL * 4);
        unsigned lbase = (unsigned)(uintptr_t)(&featbuf[buf][0]);
#pragma unroll
        for (int j = 0; j < 32; j++) {
            uint64_t off = cb + (uint64_t)(j * 32u + lane) * 16u;
            if (off > fbytes - 16) off = fbytes - 16;
            uint64_t ga = gbase + off;
            unsigned la = lbase + (unsigned)(j * 32u + lane) * 16u;
            asm volatile("global_load_async_to_lds_b128 %0, %1, off"
                         :: "v"(la), "v"(ga) : "memory");
        }
    };

    if (w0) { issue(0, 0); issue(1, 1); }

    int cur = 0;
    unsigned bpw = 0;    // 4 packed backpointer bytes
    for (int c = 0; c < NC; c++) {
        // chunks c and c+1 are in flight (32 ops each, in-order completion):
        // waiting for ASYNCcnt <= 32 guarantees chunk c has landed in LDS.
        if (w0) asm volatile("s_wait_asynccnt 32" ::: "memory");
        __syncthreads();

        const float* fb    = &featbuf[c & 1][0];
        const int    steps = imin_(CHUNK, T - c * CHUNK);

        for (int s = 0; s < steps; s++) {
            const float* vc = vbuf[cur];
            const float4 q0 = ((const float4*)vc)[0];  // ds_load_b128 broadcast
            float best = q0.x + tr[0];
            int   bpi  = 0;
            { float s1 = q0.y + tr[1]; if (s1 > best) { best = s1; bpi = 1; }
              float s2 = q0.z + tr[2]; if (s2 > best) { best = s2; bpi = 2; }
              float s3 = q0.w + tr[3]; if (s3 > best) { best = s3; bpi = 3; } }
#pragma unroll
            for (int i = 1; i < 16; i++) {
                float4 q = ((const float4*)vc)[i];
                float s0 = q.x + tr[4*i+0]; if (s0 > best) { best = s0; bpi = 4*i+0; }
                float s1 = q.y + tr[4*i+1]; if (s1 > best) { best = s1; bpi = 4*i+1; }
                float s2 = q.z + tr[4*i+2]; if (s2 > best) { best = s2; bpi = 4*i+2; }
                float s3 = q.w + tr[4*i+3]; if (s3 > best) { best = s3; bpi = 4*i+3; }
            }
            const int t = c * CHUNK + s;
            bpw |= (unsigned)bpi << (8 * (t & 3));
            if ((t & 3) == 3 || t == T - 1) {          // global_store_b32 / 4 steps
                bp32[(size_t)(t >> 2) * LBL + n] = bpw;
                bpw = 0;
            }
            vbuf[cur ^ 1][n] = best + fb[s * LBL + n];  // + emission
            cur ^= 1;
            __syncthreads();
        }
        if (w0) issue(c + 2, c & 1);   // refill the buffer chunk c just freed
    }

    // terminal = v + transitions[STOP=63][:]
    term[n] = vbuf[cur][n] + trans[63 * LBL + n];
    __syncthreads();
    if (n == 0) {
        float bv = term[0]; int bi = 0;
        for (int i = 1; i < LBL; i++)
            if (term[i] > bv) { bv = term[i]; bi = i; }   // first-max, like argmax
        *outScore = bv;
        *bestOut  = bi;
    }
}

// ---------------------------------------------------------------------------
// Kernel 2: per-block composed backpointer maps E_b(x) = f_tlo∘...∘f_thi(x).
// 64 lanes chase 64 independent chains through an LDS copy of the block's
// (packed) backpointers. Fully parallel across ~T/KBLK workgroups.
// ---------------------------------------------------------------------------
__global__ __launch_bounds__(64) void crf_blockmaps(
    const unsigned char* __restrict__ bp, unsigned char* __restrict__ E, int T)
{
    __shared__ __align__(16) unsigned char lbp[KBLK * LBL];
    const int b   = blockIdx.x;
    const int tlo = b * KBLK;
    const int kb  = imin_(KBLK, T - tlo);

    const uint4* src = (const uint4*)(bp + (size_t)tlo * LBL);
    uint4*       dst = (uint4*)lbp;
    const int nvec = (((kb + 3) >> 2) << 8) >> 4;   // ceil(kb/4)*256 bytes / 16
    for (int i = threadIdx.x; i < nvec; i += 64) dst[i] = src[i];
    __syncthreads();

    int tag = threadIdx.x;
    for (int i = kb - 1; i >= 0; --i) tag = lbp[bpLdsIdx(i, tag)];
    E[(size_t)b * LBL + threadIdx.x] = (unsigned char)tag;
}

// ---------------------------------------------------------------------------
// Kernel 3: tiny sequential scan over block maps (B ~ 391 steps).
// x[b] = path value at the last timestep of block b.
// ---------------------------------------------------------------------------
__global__ void crf_scan(const unsigned char* __restrict__ E,
                         const int* __restrict__ bestIn,
                         int* __restrict__ x, int B)
{
    if (blockIdx.x == 0 && threadIdx.x == 0) {
        int tag = *bestIn;
        x[B - 1] = tag;
        for (int b = B - 1; b >= 1; --b) {
            tag = E[(size_t)b * LBL + tag];
            x[b - 1] = tag;
        }
    }
}

// ---------------------------------------------------------------------------
// Kernel 4: per-block path emission. Lane 0 re-chases the single winning chain
// through LDS; all 64 lanes then write the tags out as floats (coalesced).
// ---------------------------------------------------------------------------
__global__ __launch_bounds__(64) void crf_emit(
    const unsigned char* __restrict__ bp, const int* __restrict__ x,
    float* __restrict__ path, int T)
{
    __shared__ __align__(16) unsigned char lbp[KBLK * LBL];
    __shared__ unsigned char tags[KBLK];
    const int b   = blockIdx.x;
    const int tlo = b * KBLK;
    const int kb  = imin_(KBLK, T - tlo);

    const uint4* src = (const uint4*)(bp + (size_t)tlo * LBL);
    uint4*       dst = (uint4*)lbp;
    const int nvec = (((kb + 3) >> 2) << 8) >> 4;
    for (int i = threadIdx.x; i < nvec; i += 64) dst[i] = src[i];
    __syncthreads();

    if (threadIdx.x == 0) {
        int tag = x[b];
        tags[kb - 1] = (unsigned char)tag;          // path at t_hi
        for (int i = kb - 1; i >= 1; --i) {         // path[t-1] = bp[t][path[t]]
            tag = lbp[bpLdsIdx(i, tag)];
            tags[i - 1] = (unsigned char)tag;
        }
    }
    __syncthreads();
    for (int i = threadIdx.x; i < kb; i += 64)
        path[tlo + i] = (float)tags[i];
}

// ---------------------------------------------------------------------------
extern "C" void kernel_launch(void* const* d_in, const int* in_sizes, int n_in,
                              void* d_out, int out_size, void* d_ws, size_t ws_size,
                              hipStream_t stream)
{
    const float* feats = (const float*)d_in[0];   // (1,T,64) f32
    const float* trans = (const float*)d_in[1];   // (64,64)  f32
    const int T = in_sizes[0] / LBL;
    const int B = (T + KBLK - 1) / KBLK;

    float* out = (float*)d_out;                   // [0]=score, [1..T]=path

    // workspace layout (256-aligned regions)
    unsigned char* bp = (unsigned char*)d_ws;                 // ceil(T/4)*256 bytes
    size_t off = ((size_t)T * LBL + 255) & ~(size_t)255;
    unsigned char* E = (unsigned char*)d_ws + off;            // B*64 bytes
    off += (((size_t)B * LBL) + 255) & ~(size_t)255;
    int* xarr = (int*)((char*)d_ws + off);                    // B ints
    off += (((size_t)B * 4) + 255) & ~(size_t)255;
    int* bestp = (int*)((char*)d_ws + off);                   // 1 int

    crf_forward  <<<1, 64, 0, stream>>>(feats, trans, (unsigned*)bp, out, bestp, T);
    crf_blockmaps<<<B, 64, 0, stream>>>(bp, E, T);
    crf_scan     <<<1, 1, 0, stream>>>(E, bestp, xarr, B);
    crf_emit     <<<B, 64, 0, stream>>>(bp, xarr, out + 1, T);
}